// Eghnv2Policy_38448547234232
// MI455X (gfx1250) — compile-verified
//
#include <hip/hip_runtime.h>

typedef __attribute__((ext_vector_type(2))) float v2f;
typedef __attribute__((ext_vector_type(8))) float v8f;

#define NAG 12           // agents per env
#define NENV 4096        // envs
#define HID 64
#define INVNF 32
#define NVEC 4
#define DIMS 3
#define EDGES 132        // 12*11
#define KIN 132          // 2*HID + NVEC
#define BLOCK 96         // 3 wave32s: each wave owns 3 16-edge M-tiles
#define MTPW 3           // M-tiles per wave (B-fragment reuse factor)
#define MST 68           // LDS row stride (floats) -> conflict-free A-fragment loads

__device__ __forceinline__ float silu_f(float x) {
  return x * (1.0f / (1.0f + __expf(-x)));
}

// D = A(16x4 f32) * B(4x16 f32) + C(16x16 f32)
__device__ __forceinline__ v8f wmma4(v2f a, v2f b, v8f c) {
  return __builtin_amdgcn_wmma_f32_16x16x4_f32(false, a, false, b, (short)0, c,
                                               false, false);
}

__global__ __launch_bounds__(BLOCK)
void egnn_policy_kernel(const float* __restrict__ h0,
                        const float* __restrict__ x0,
                        const float* __restrict__ W_embed,
                        const float* __restrict__ b_embed,
                        const float* __restrict__ We1,
                        const float* __restrict__ be1,
                        const float* __restrict__ We2,
                        const float* __restrict__ be2,
                        const float* __restrict__ Wx,
                        const float* __restrict__ bx,
                        const float* __restrict__ Wh1,
                        const float* __restrict__ bh1,
                        const float* __restrict__ Wh2,
                        const float* __restrict__ bh2,
                        const float* __restrict__ w_act,
                        const float* __restrict__ log_std,
                        float* __restrict__ out) {
  __shared__ float sh_h[16 * MST];    // node embeddings, rows 12..15 = 0
  __shared__ float sh_agg[16 * MST];  // aggregated messages, rows 12..15 = 0
  __shared__ float sh_m[144 * MST];   // edge messages (m1 staging, then m2); node staging
  __shared__ float sh_xa[NAG * 12];   // x buffer A  [node][d*4+v]
  __shared__ float sh_xb[NAG * 12];   // x buffer B
  __shared__ float sh_rw[144 * 4];    // radial, then wc

  const int tid = threadIdx.x;
  const int lane = tid & 31;
  const int wave = tid >> 5;           // 0..2; wave owns tiles 3w..3w+2
  const int l15 = lane & 15;
  const int kh = (lane < 16) ? 0 : 2;  // K split of A/B fragments at lane 16
  const int env = blockIdx.x;

  // edge identities for this wave's 3 tiles (same for lanes l and l+16)
  int eA[MTPW], eiA[MTPW], ejA[MTPW];
#pragma unroll
  for (int s = 0; s < MTPW; ++s) {
    int ee = 16 * (MTPW * wave + s) + l15;
    eA[s] = ee;
    if (ee < EDGES) {
      int ii = ee / 11;
      int jj = ee - ii * 11;
      eiA[s] = ii;
      ejA[s] = jj + (jj >= ii);
    } else {  // pad edges: computed but never aggregated
      eiA[s] = 0;
      ejA[s] = 1;
    }
  }

  // ---------------- init ----------------
  for (int idx = tid; idx < 16 * MST; idx += BLOCK) {
    sh_agg[idx] = 0.0f;
    if ((idx / MST) >= NAG) sh_h[idx] = 0.0f;  // zero pad rows of h
  }
  for (int idx = tid; idx < NAG * 12; idx += BLOCK)
    sh_xa[idx] = x0[env * NAG * 12 + idx];

  // ---------------- embedding: h = h0 @ W_embed + b_embed (wave 0) -------
  if (wave == 0) {
    v8f acc[4];
#pragma unroll
    for (int nt = 0; nt < 4; ++nt) {
      float bv = b_embed[nt * 16 + l15];
#pragma unroll
      for (int r = 0; r < 8; ++r) acc[nt][r] = bv;
    }
    const int arow = (l15 < NAG) ? l15 : 0;  // clamp; garbage rows discarded
    const float* arp = h0 + (env * NAG + arow) * INVNF;
#pragma unroll
    for (int kk = 0; kk < 8; ++kk) {
      int k0 = 4 * kk;
      v2f a;
      a.x = arp[k0 + kh];
      a.y = arp[k0 + kh + 1];
#pragma unroll
      for (int nt = 0; nt < 4; ++nt) {
        v2f b;
        b.x = W_embed[(k0 + kh) * HID + nt * 16 + l15];
        b.y = W_embed[(k0 + kh + 1) * HID + nt * 16 + l15];
        acc[nt] = wmma4(a, b, acc[nt]);
      }
    }
    const int rb = (lane < 16) ? 0 : 8;
#pragma unroll
    for (int nt = 0; nt < 4; ++nt)
#pragma unroll
      for (int r = 0; r < 8; ++r) {
        int row = rb + r;
        if (row < NAG) sh_h[row * MST + nt * 16 + l15] = acc[nt][r];
      }
  }
  __syncthreads();

  // ---------------- layers ----------------
  for (int l = 0; l < 2; ++l) {
    const float* We1_l = We1 + l * KIN * HID;
    const float* be1_l = be1 + l * HID;
    const float* We2_l = We2 + l * HID * HID;
    const float* be2_l = be2 + l * HID;
    const float* Wx_l = Wx + l * HID * NVEC;
    const float* bx_l = bx + l * NVEC;
    const float* Wh1_l = Wh1 + l * 2 * HID * HID;
    const float* bh1_l = bh1 + l * HID;
    const float* Wh2_l = Wh2 + l * HID * HID;
    const float* bh2_l = bh2 + l * HID;
    float* xin = (l == 0) ? sh_xa : sh_xb;
    float* xout = (l == 0) ? sh_xb : sh_xa;

    // ---- radials for this wave's tiles (wave-local; consumed only by self)
    if (lane < 16) {
#pragma unroll
      for (int s = 0; s < MTPW; ++s) {
        int ii = eiA[s], jj = ejA[s];
        float r0 = 0, r1 = 0, r2 = 0, r3 = 0;
#pragma unroll
        for (int d = 0; d < DIMS; ++d) {
          float d0 = xin[ii * 12 + d * 4 + 0] - xin[jj * 12 + d * 4 + 0];
          float d1 = xin[ii * 12 + d * 4 + 1] - xin[jj * 12 + d * 4 + 1];
          float d2 = xin[ii * 12 + d * 4 + 2] - xin[jj * 12 + d * 4 + 2];
          float d3 = xin[ii * 12 + d * 4 + 3] - xin[jj * 12 + d * 4 + 3];
          r0 += d0 * d0; r1 += d1 * d1; r2 += d2 * d2; r3 += d3 * d3;
        }
        sh_rw[eA[s] * 4 + 0] = r0;
        sh_rw[eA[s] * 4 + 1] = r1;
        sh_rw[eA[s] * 4 + 2] = r2;
        sh_rw[eA[s] * 4 + 3] = r3;
      }
    }

    // ---- m1 = silu([h_i | h_j | radial] @ We1 + be1) ----
    // 33 ksteps x 4 ntiles x 3 Mtiles; each B fragment reused 3x.
    v8f acc[MTPW][4];
#pragma unroll
    for (int nt = 0; nt < 4; ++nt) {
      float bv = be1_l[nt * 16 + l15];
#pragma unroll
      for (int s = 0; s < MTPW; ++s)
#pragma unroll
        for (int r = 0; r < 8; ++r) acc[s][nt][r] = bv;
    }
    for (int kk = 0; kk < 33; ++kk) {
      int k0 = 4 * kk;
      v2f a[MTPW];
#pragma unroll
      for (int s = 0; s < MTPW; ++s) {
        if (kk < 16) {
          a[s].x = sh_h[eiA[s] * MST + k0 + kh];
          a[s].y = sh_h[eiA[s] * MST + k0 + kh + 1];
        } else if (kk < 32) {
          a[s].x = sh_h[ejA[s] * MST + (k0 - 64) + kh];
          a[s].y = sh_h[ejA[s] * MST + (k0 - 64) + kh + 1];
        } else {
          a[s].x = sh_rw[eA[s] * 4 + kh];
          a[s].y = sh_rw[eA[s] * 4 + kh + 1];
        }
      }
#pragma unroll
      for (int nt = 0; nt < 4; ++nt) {
        v2f b;
        b.x = We1_l[(k0 + kh) * HID + nt * 16 + l15];
        b.y = We1_l[(k0 + kh + 1) * HID + nt * 16 + l15];
#pragma unroll
        for (int s = 0; s < MTPW; ++s) acc[s][nt] = wmma4(a[s], b, acc[s][nt]);
      }
    }
    {  // silu + stage to LDS (needed as A-matrix for m2)
#pragma unroll
      for (int s = 0; s < MTPW; ++s) {
        const int rb = 16 * (MTPW * wave + s) + ((lane < 16) ? 0 : 8);
#pragma unroll
        for (int nt = 0; nt < 4; ++nt)
#pragma unroll
          for (int r = 0; r < 8; ++r)
            sh_m[(rb + r) * MST + nt * 16 + l15] = silu_f(acc[s][nt][r]);
      }
    }

    // ---- m2 = silu(m1 @ We2 + be2) : 16 ksteps x 4 ntiles x 3 Mtiles ----
#pragma unroll
    for (int nt = 0; nt < 4; ++nt) {
      float bv = be2_l[nt * 16 + l15];
#pragma unroll
      for (int s = 0; s < MTPW; ++s)
#pragma unroll
        for (int r = 0; r < 8; ++r) acc[s][nt][r] = bv;
    }
    for (int kk = 0; kk < 16; ++kk) {
      int k0 = 4 * kk;
      v2f a[MTPW];
#pragma unroll
      for (int s = 0; s < MTPW; ++s) {
        int arow = 16 * (MTPW * wave + s) + l15;
        a[s].x = sh_m[arow * MST + k0 + kh];
        a[s].y = sh_m[arow * MST + k0 + kh + 1];
      }
#pragma unroll
      for (int nt = 0; nt < 4; ++nt) {
        v2f b;
        b.x = We2_l[(k0 + kh) * HID + nt * 16 + l15];
        b.y = We2_l[(k0 + kh + 1) * HID + nt * 16 + l15];
#pragma unroll
        for (int s = 0; s < MTPW; ++s) acc[s][nt] = wmma4(a[s], b, acc[s][nt]);
      }
    }
    {  // silu + overwrite staging with final m2
#pragma unroll
      for (int s = 0; s < MTPW; ++s) {
        const int rb = 16 * (MTPW * wave + s) + ((lane < 16) ? 0 : 8);
#pragma unroll
        for (int nt = 0; nt < 4; ++nt)
#pragma unroll
          for (int r = 0; r < 8; ++r)
            sh_m[(rb + r) * MST + nt * 16 + l15] = silu_f(acc[s][nt][r]);
      }
    }
    __syncthreads();

    // ---- segment sums: agg[i][c] = sum of node i's 11 contiguous edges
    for (int item = tid; item < NAG * HID; item += BLOCK) {
      int i = item >> 6, c = item & 63;
      float s = 0.0f;
#pragma unroll
      for (int k = 0; k < 11; ++k) s += sh_m[(11 * i + k) * MST + c];
      sh_agg[i * MST + c] = s;
    }
    // ---- wc = m2 @ Wx + bx (2% of FLOPs -> VALU), reuse sh_rw
    for (int item = tid; item < EDGES * NVEC; item += BLOCK) {
      int ew = item >> 2, v = item & 3;
      float s = bx_l[v];
#pragma unroll
      for (int c = 0; c < HID; ++c) s += sh_m[ew * MST + c] * Wx_l[c * 4 + v];
      sh_rw[ew * 4 + v] = s;
    }
    __syncthreads();

    // ---- coordinate update (waves 1..2), concurrent with node MLP
    if (wave != 0) {
      for (int item = tid - 32; item < NAG * 12; item += 64) {
        int i = item / 12, rem = item - 12 * i;  // rem = d*4+v
        int v = rem & 3;
        float xi = xin[i * 12 + rem];
        float s = 0.0f;
#pragma unroll
        for (int k = 0; k < 11; ++k) {
          int ee = 11 * i + k;
          int j = k + (k >= i);
          s += (xi - xin[j * 12 + rem]) * sh_rw[ee * 4 + v];
        }
        xout[i * 12 + rem] = xi + s * (1.0f / 11.0f);
      }
    }

    // ---- node MLP (wave 0): h += silu([h|agg]@Wh1+bh1)@Wh2 + bh2
    if (wave == 0) {
      v8f nacc[4];
#pragma unroll
      for (int nt = 0; nt < 4; ++nt) {
        float bv = bh1_l[nt * 16 + l15];
#pragma unroll
        for (int r = 0; r < 8; ++r) nacc[nt][r] = bv;
      }
      for (int kk = 0; kk < 32; ++kk) {
        int k0 = 4 * kk;
        v2f a;
        if (kk < 16) {
          a.x = sh_h[l15 * MST + k0 + kh];
          a.y = sh_h[l15 * MST + k0 + kh + 1];
        } else {
          a.x = sh_agg[l15 * MST + (k0 - 64) + kh];
          a.y = sh_agg[l15 * MST + (k0 - 64) + kh + 1];
        }
#pragma unroll
        for (int nt = 0; nt < 4; ++nt) {
          v2f b;
          b.x = Wh1_l[(k0 + kh) * HID + nt * 16 + l15];
          b.y = Wh1_l[(k0 + kh + 1) * HID + nt * 16 + l15];
          nacc[nt] = wmma4(a, b, nacc[nt]);
        }
      }
      {  // stage silu(u) into sh_m rows 0..15 (free after barrier)
        const int rb = (lane < 16) ? 0 : 8;
#pragma unroll
        for (int nt = 0; nt < 4; ++nt)
#pragma unroll
          for (int r = 0; r < 8; ++r)
            sh_m[(rb + r) * MST + nt * 16 + l15] = silu_f(nacc[nt][r]);
      }
#pragma unroll
      for (int nt = 0; nt < 4; ++nt) {
        float bv = bh2_l[nt * 16 + l15];
#pragma unroll
        for (int r = 0; r < 8; ++r) nacc[nt][r] = bv;
      }
      for (int kk = 0; kk < 16; ++kk) {
        int k0 = 4 * kk;
        v2f a;
        a.x = sh_m[l15 * MST + k0 + kh];
        a.y = sh_m[l15 * MST + k0 + kh + 1];
#pragma unroll
        for (int nt = 0; nt < 4; ++nt) {
          v2f b;
          b.x = Wh2_l[(k0 + kh) * HID + nt * 16 + l15];
          b.y = Wh2_l[(k0 + kh + 1) * HID + nt * 16 + l15];
          nacc[nt] = wmma4(a, b, nacc[nt]);
        }
      }
      {  // h += delta (valid rows only; pad rows stay 0)
        const int rb = (lane < 16) ? 0 : 8;
#pragma unroll
        for (int nt = 0; nt < 4; ++nt)
#pragma unroll
          for (int r = 0; r < 8; ++r) {
            int row = rb + r;
            if (row < NAG) sh_h[row * MST + nt * 16 + l15] += nacc[nt][r];
          }
      }
    }
    __syncthreads();
  }

  // ---------------- outputs ----------------
  // actions[env][i][d] = sum_v x[i][d][v]*w_act[v]; log_probs constant.
  if (tid < NAG * DIMS) {
    int i = tid / 3, d = tid - 3 * i;
    float mu = 0.0f;
#pragma unroll
    for (int v = 0; v < NVEC; ++v) mu += sh_xa[i * 12 + d * 4 + v] * w_act[v];
    out[env * (NAG * DIMS) + tid] = mu;
    out[NENV * NAG * DIMS + env * (NAG * DIMS) + tid] =
        -log_std[d] - 0.91893853320467274f;  // 0.5*ln(2*pi)
  }
}

extern "C" void kernel_launch(void* const* d_in, const int* in_sizes, int n_in,
                              void* d_out, int out_size, void* d_ws,
                              size_t ws_size, hipStream_t stream) {
  (void)in_sizes; (void)n_in; (void)d_ws; (void)ws_size; (void)out_size;
  const float* h0 = (const float*)d_in[0];
  const float* x0 = (const float*)d_in[1];
  const float* W_embed = (const float*)d_in[2];
  const float* b_embed = (const float*)d_in[3];
  const float* We1 = (const float*)d_in[4];
  const float* be1 = (const float*)d_in[5];
  const float* We2 = (const float*)d_in[6];
  const float* be2 = (const float*)d_in[7];
  const float* Wx = (const float*)d_in[8];
  const float* bx = (const float*)d_in[9];
  const float* Wh1 = (const float*)d_in[10];
  const float* bh1 = (const float*)d_in[11];
  const float* Wh2 = (const float*)d_in[12];
  const float* bh2 = (const float*)d_in[13];
  const float* w_act = (const float*)d_in[14];
  const float* log_std = (const float*)d_in[15];
  // d_in[16]=row, d_in[17]=col: static topology, recomputed in-kernel.
  float* out = (float*)d_out;

  egnn_policy_kernel<<<NENV, BLOCK, 0, stream>>>(
      h0, x0, W_embed, b_embed, We1, be1, We2, be2, Wx, bx, Wh1, bh1, Wh2, bh2,
      w_act, log_std, out);
}